// Block_28192165331060
// MI455X (gfx1250) — compile-verified
//
#include <hip/hip_runtime.h>
#include <math.h>

typedef __attribute__((ext_vector_type(2))) float v2f;
typedef __attribute__((ext_vector_type(8))) float v8f;

#define DM 128
#define MT 4   // M sub-tiles (of 16 rows) per block

// ---------------------------------------------------------------------------
// float atomic max via sign-split int atomics (buffer must be inited to -inf)
// ---------------------------------------------------------------------------
__device__ __forceinline__ void atomicMaxF(float* addr, float val) {
  if (val >= 0.0f) atomicMax((int*)addr, __float_as_int(val));
  else             atomicMin((unsigned int*)addr, __float_as_uint(val));
}

// ---------------------------------------------------------------------------
// init: m = -inf, den = 0, BN accumulators (512 floats) = 0
// ---------------------------------------------------------------------------
__global__ void init_kernel(float* __restrict__ m, float* __restrict__ den,
                            float* __restrict__ sums, int N) {
  int i = blockIdx.x * blockDim.x + threadIdx.x;
  if (i < N) { m[i] = -INFINITY; den[i] = 0.0f; }
  if (i < 512) sums[i] = 0.0f;
}

// ---------------------------------------------------------------------------
// fp32 WMMA GEMM:  C[M x Dn] = A[M x K] @ W[K x Dn] + bias (+relu) (+add)
// block = 128 threads (4 waves); block tile = 64(M) x 128(N); grid.y = Dn/128
// A tile (64 x K) staged in LDS, padded ldk = K+4 (kills bank conflicts).
// Each wave: 4 M-subtiles x 2 N-tiles = 8 accumulators; per K-step (K+=4)
// it loads 2 B fragments and fires 8 V_WMMA_F32_16X16X4_F32 against them,
// amortizing weight-fragment traffic 4x vs a 16-row tile.
// ---------------------------------------------------------------------------
__global__ void __launch_bounds__(128) gemm_wmma_kernel(
    const float* __restrict__ A, const float* __restrict__ W,
    const float* __restrict__ bias, const float* __restrict__ add,
    float* __restrict__ C, int M, int K, int Dn, int relu) {
  __shared__ float As[64 * 260];           // 64 rows, supports K up to 256
  const int ldk  = K + 4;
  const int tid  = threadIdx.x;
  const int wave = tid >> 5;
  const int lane = tid & 31;
  const int m0   = blockIdx.x << 6;        // 64 rows per block
  const int n0   = blockIdx.y * 128 + wave * 32;

  // cooperative, coalesced load of the 64 x K A tile (row-clamped at edge)
  const int tileK = 64 * K;
  for (int idx = tid * 4; idx < tileK; idx += 512) {
    int r = idx / K;
    int c = idx - r * K;
    int row = m0 + r;
    row = (row < M) ? row : (M - 1);
    float4 t = *(const float4*)(A + (size_t)row * K + c);
    *(float4*)(&As[r * ldk + c]) = t;
  }
  __syncthreads();

  v8f acc[MT][2];
  for (int mt = 0; mt < MT; ++mt)
    for (int j = 0; j < 2; ++j)
      acc[mt][j] = (v8f){0.f, 0.f, 0.f, 0.f, 0.f, 0.f, 0.f, 0.f};

  const int arow = lane & 15;          // A: lanes 0-15 rows, repeated 16-31
  const int koff = (lane >> 4) << 1;   // A/B: lanes 16-31 hold K=2,3
  const int ncol = lane & 15;          // B/C: column within tile

  for (int k0 = 0; k0 < K; k0 += 4) {
    const float* Wp = W + (size_t)(k0 + koff) * Dn;
    v2f b0, b1;
    b0.x = Wp[n0 + ncol];
    b0.y = Wp[Dn + n0 + ncol];
    b1.x = Wp[n0 + 16 + ncol];
    b1.y = Wp[Dn + n0 + 16 + ncol];
    const float* Ap = &As[arow * ldk + k0 + koff];
#pragma unroll
    for (int mt = 0; mt < MT; ++mt) {
      v2f a = *(const v2f*)(Ap + mt * 16 * ldk);
      acc[mt][0] = __builtin_amdgcn_wmma_f32_16x16x4_f32(
          false, a, false, b0, (short)0, acc[mt][0], false, false);
      acc[mt][1] = __builtin_amdgcn_wmma_f32_16x16x4_f32(
          false, a, false, b1, (short)0, acc[mt][1], false, false);
    }
  }

  // C layout: VGPR r -> row r (lanes 0-15) / row r+8 (lanes 16-31)
  const int rbase = (lane >> 4) << 3;
  const float bia0 = bias[n0 + ncol];
  const float bia1 = bias[n0 + 16 + ncol];
  for (int mt = 0; mt < MT; ++mt) {
    const int mrow = m0 + mt * 16 + rbase;
    for (int r = 0; r < 8; ++r) {
      int mm = mrow + r;
      if (mm >= M) continue;
      size_t o0 = (size_t)mm * Dn + n0 + ncol;
      size_t o1 = o0 + 16;
      float v0 = acc[mt][0][r] + bia0;
      float v1 = acc[mt][1][r] + bia1;
      if (relu) { v0 = fmaxf(v0, 0.f); v1 = fmaxf(v1, 0.f); }
      if (add)  { v0 += add[o0];       v1 += add[o1]; }
      C[o0] = v0;
      C[o1] = v1;
    }
  }
}

// ---------------------------------------------------------------------------
// wave per edge: score = dot(q[dst], k[src]) / sqrt(128); segment max into m
// ---------------------------------------------------------------------------
__global__ void edge_scores_kernel(const float* __restrict__ q,
                                   const float* __restrict__ k,
                                   const int* __restrict__ src,
                                   const int* __restrict__ dst,
                                   float* __restrict__ scores,
                                   float* __restrict__ m, int E) {
  int e = (int)((blockIdx.x * (size_t)blockDim.x + threadIdx.x) >> 5);
  int lane = threadIdx.x & 31;
  if (e >= E) return;
  int s = src[e], d = dst[e];
  float4 qa = ((const float4*)(q + (size_t)d * DM))[lane];
  float4 kb = ((const float4*)(k + (size_t)s * DM))[lane];
  float val = qa.x * kb.x + qa.y * kb.y + qa.z * kb.z + qa.w * kb.w;
  for (int o = 16; o > 0; o >>= 1) val += __shfl_xor(val, o, 32);
  val *= 0.08838834764831845f;  // 1/sqrt(128)
  if (lane == 0) {
    scores[e] = val;
    atomicMaxF(&m[d], val);
  }
}

// ---------------------------------------------------------------------------
// thread per edge: p = exp(score - m[dst]); segment sum into den
// ---------------------------------------------------------------------------
__global__ void edge_exp_kernel(const float* __restrict__ scores,
                                const int* __restrict__ dst,
                                const float* __restrict__ m,
                                float* __restrict__ p,
                                float* __restrict__ den, int E) {
  int e = blockIdx.x * blockDim.x + threadIdx.x;
  if (e >= E) return;
  int d = dst[e];
  float pe = expf(scores[e] - m[d]);
  p[e] = pe;
  atomicAdd(&den[d], pe);
}

// ---------------------------------------------------------------------------
// wave per edge: agg[dst] += (p/den[dst]) * v[src]   (agg pre-inited to skip)
// ---------------------------------------------------------------------------
__global__ void edge_agg_kernel(const float* __restrict__ v,
                                const int* __restrict__ src,
                                const int* __restrict__ dst,
                                const float* __restrict__ p,
                                const float* __restrict__ den,
                                float* __restrict__ agg, int E) {
  int e = (int)((blockIdx.x * (size_t)blockDim.x + threadIdx.x) >> 5);
  int lane = threadIdx.x & 31;
  if (e >= E) return;
  int s = src[e], d = dst[e];
  float alpha = p[e] / den[d];
  float4 vv = ((const float4*)(v + (size_t)s * DM))[lane];
  float* base = agg + (size_t)d * DM + lane * 4;
  atomicAdd(base + 0, alpha * vv.x);
  atomicAdd(base + 1, alpha * vv.y);
  atomicAdd(base + 2, alpha * vv.z);
  atomicAdd(base + 3, alpha * vv.w);
}

// ---------------------------------------------------------------------------
// BN column stats: sums[0..127] = sum, sums[128..255] = sumsq
// ---------------------------------------------------------------------------
__global__ void __launch_bounds__(256) bn_stats_kernel(
    const float* __restrict__ h, float* __restrict__ sums, int N) {
  __shared__ float ssum[256], ssq[256];
  int col  = threadIdx.x & 127;
  int half = threadIdx.x >> 7;
  float s = 0.f, sq = 0.f;
  for (int r = blockIdx.x * 2 + half; r < N; r += gridDim.x * 2) {
    float x = h[(size_t)r * DM + col];
    s += x;
    sq += x * x;
  }
  ssum[threadIdx.x] = s;
  ssq[threadIdx.x]  = sq;
  __syncthreads();
  if (threadIdx.x < 128) {
    atomicAdd(&sums[col],       ssum[col] + ssum[col + 128]);
    atomicAdd(&sums[128 + col], ssq[col]  + ssq[col + 128]);
  }
}

// ---------------------------------------------------------------------------
// BN apply: out = g*(h-mu)/sqrt(var+eps) + b  (biased var from sums)
// ---------------------------------------------------------------------------
__global__ void bn_apply_kernel(const float* __restrict__ h,
                                const float* __restrict__ sums,
                                const float* __restrict__ g,
                                const float* __restrict__ b,
                                float* __restrict__ out, int N) {
  size_t i = (size_t)blockIdx.x * blockDim.x + threadIdx.x;
  if (i >= (size_t)N * DM) return;
  int col = (int)(i & (DM - 1));
  float inv = 1.0f / (float)N;
  float mu  = sums[col] * inv;
  float var = sums[128 + col] * inv - mu * mu;
  out[i] = g[col] * (h[i] - mu) / sqrtf(var + 1e-5f) + b[col];
}

// ---------------------------------------------------------------------------
extern "C" void kernel_launch(void* const* d_in, const int* in_sizes, int n_in,
                              void* d_out, int out_size, void* d_ws, size_t ws_size,
                              hipStream_t stream) {
  const float* x  = (const float*)d_in[0];
  const int*   ei = (const int*)d_in[1];
  const float* Wq = (const float*)d_in[2];  const float* bq = (const float*)d_in[3];
  const float* Wk = (const float*)d_in[4];  const float* bk = (const float*)d_in[5];
  const float* Wv = (const float*)d_in[6];  const float* bv = (const float*)d_in[7];
  const float* Ws = (const float*)d_in[8];  const float* bs = (const float*)d_in[9];
  const float* WO = (const float*)d_in[10]; const float* bO = (const float*)d_in[11];
  const float* W1 = (const float*)d_in[12]; const float* b1 = (const float*)d_in[13];
  const float* W2 = (const float*)d_in[14]; const float* b2 = (const float*)d_in[15];
  const float* g1 = (const float*)d_in[16]; const float* be1 = (const float*)d_in[17];
  const float* g2 = (const float*)d_in[18]; const float* be2 = (const float*)d_in[19];

  const int N = in_sizes[0] / DM;   // 50000
  const int E = in_sizes[1] / 2;    // 800000
  const int* src = ei;
  const int* dst = ei + E;

  float* ws = (float*)d_ws;
  const size_t ND = (size_t)N * DM;
  float* q      = ws;              // later reused as h / h2
  float* kk     = ws + ND;         // later reused as h_bn1
  float* v      = ws + 2 * ND;
  float* agg    = ws + 3 * ND;     // inited with skip projection
  float* t      = ws + 2 * ND;     // FFN hidden [N x 256], reuses v+agg
  float* scores = ws + 4 * ND;
  float* p      = scores + E;
  float* m      = p + E;
  float* den    = m + N;
  float* sums1  = den + N;         // 256 floats
  float* sums2  = sums1 + 256;     // 256 floats

  const int MB = (N + 63) / 64;    // 782 row-tiles of 64

  init_kernel<<<(N + 255) / 256, 256, 0, stream>>>(m, den, sums1, N);

  // q, k, v, skip(->agg)
  gemm_wmma_kernel<<<dim3(MB, 1), 128, 0, stream>>>(x, Wq, bq, nullptr, q,   N, DM, DM, 0);
  gemm_wmma_kernel<<<dim3(MB, 1), 128, 0, stream>>>(x, Wk, bk, nullptr, kk,  N, DM, DM, 0);
  gemm_wmma_kernel<<<dim3(MB, 1), 128, 0, stream>>>(x, Wv, bv, nullptr, v,   N, DM, DM, 0);
  gemm_wmma_kernel<<<dim3(MB, 1), 128, 0, stream>>>(x, Ws, bs, nullptr, agg, N, DM, DM, 0);

  // edge softmax + aggregation
  edge_scores_kernel<<<(E + 7) / 8, 256, 0, stream>>>(q, kk, src, dst, scores, m, E);
  edge_exp_kernel<<<(E + 255) / 256, 256, 0, stream>>>(scores, dst, m, p, den, E);
  edge_agg_kernel<<<(E + 7) / 8, 256, 0, stream>>>(v, src, dst, p, den, agg, E);

  // h = (agg + skip) @ WO + bO + x     (writes into q region)
  gemm_wmma_kernel<<<dim3(MB, 1), 128, 0, stream>>>(agg, WO, bO, x, q, N, DM, DM, 0);

  // BN1 -> h_bn1 (kk region)
  bn_stats_kernel<<<256, 256, 0, stream>>>(q, sums1, N);
  bn_apply_kernel<<<(int)((ND + 255) / 256), 256, 0, stream>>>(q, sums1, g1, be1, kk, N);

  // FFN: t = relu(h_bn1 @ W1 + b1);  h2 = t @ W2 + b2 + h_bn1 (-> q region)
  gemm_wmma_kernel<<<dim3(MB, 2), 128, 0, stream>>>(kk, W1, b1, nullptr, t, N, DM, 2 * DM, 1);
  gemm_wmma_kernel<<<dim3(MB, 1), 128, 0, stream>>>(t, W2, b2, kk, q, N, 2 * DM, DM, 0);

  // BN2 -> d_out
  bn_stats_kernel<<<256, 256, 0, stream>>>(q, sums2, N);
  bn_apply_kernel<<<(int)((ND + 255) / 256), 256, 0, stream>>>(q, sums2, g2, be2,
                                                               (float*)d_out, N);
}